// myGATConv_s_attn2_13975823582069
// MI455X (gfx1250) — compile-verified
//
#include <hip/hip_runtime.h>
#include <hip/hip_bf16.h>
#include <math.h>

// ---------------------------------------------------------------------------
// GATConv forward for MI455X (gfx1250, wave32).
// Memory-bound (~1 GB traffic, ~15 GFLOP): fuse edge-GEMM + attention dot,
// V_WMMA_F32_16X16X4_F32 for both GEMMs, ASYNC global->LDS tile staging
// (gfx1250 GLOBAL_LOAD_ASYNC_TO_LDS_B128, ASYNCcnt), NT hint on the one-shot
// e_feat stream so feat_src + softmax accumulators stay L2-resident,
// L2 atomics for the segment softmax / segment sums.
// ---------------------------------------------------------------------------

typedef __attribute__((ext_vector_type(2))) float v2f;
typedef __attribute__((ext_vector_type(8))) float v8f;

#define NNODES 50000
#define NEDGES 1600000
#define INFEAT 256
#define EFEAT  64
#define NH     4
#define HD     16
#define OUTF   64   // NH*HD

// -------- order-preserving float<->uint key for atomicMax segment-max -------
__device__ __forceinline__ unsigned fkey(float f) {
    unsigned u = __float_as_uint(f);
    return (u & 0x80000000u) ? ~u : (u | 0x80000000u);
}
__device__ __forceinline__ float funkey(unsigned k) {
    unsigned u = (k & 0x80000000u) ? (k & 0x7FFFFFFFu) : ~k;
    return __uint_as_float(u);
}

// -------- gfx1250 async global->LDS copy (ASYNCcnt-tracked, no VGPR hop) ----
// GVS mode: mem = SADDR(64b sgpr pair) + VADDR(32b byte off); VDST = LDS addr.
__device__ __forceinline__ void async_copy_b128(unsigned lds_byte_addr,
                                                const void* sbase,
                                                unsigned byte_off) {
    asm volatile("global_load_async_to_lds_b128 %0, %1, %2"
                 :: "v"(lds_byte_addr), "v"(byte_off),
                    "s"((unsigned long long)sbase)
                 : "memory");
}
__device__ __forceinline__ void async_copy_b128_nt(unsigned lds_byte_addr,
                                                   const void* sbase,
                                                   unsigned byte_off) {
    asm volatile("global_load_async_to_lds_b128 %0, %1, %2 th:TH_LOAD_NT"
                 :: "v"(lds_byte_addr), "v"(byte_off),
                    "s"((unsigned long long)sbase)
                 : "memory");
}
__device__ __forceinline__ void async_wait0() {
    asm volatile("s_wait_asynccnt 0x0" ::: "memory");
}

// ---------------------------------------------------------------------------
// K1: feat_src = feat @ fc_W.T   [N,256]x[256,64] -> [N,64]
// block = 128 threads = 4 waves; one 16-node row tile per block,
// wave w computes the 16-col tile w. K-loop over 256 in steps of 4 (WMMA x64).
// A tile staged via async-to-LDS (padded pitch 260: 260 % 64 == 4 -> no bank
// conflicts on the 16-lane column reads; 260*4 bytes is 16B-aligned).
// ---------------------------------------------------------------------------
__global__ __launch_bounds__(128)
void k_node_gemm(const float* __restrict__ feat,
                 const float* __restrict__ fcW,
                 float* __restrict__ feat_src) {
    __shared__ float As[16 * 260];
    const int tile = blockIdx.x;
    const int tid  = threadIdx.x;
    const int wave = tid >> 5;
    const int lane = tid & 31;
    const int half = lane >> 4;
    const int l16  = lane & 15;

    // async-stage the 16x256 A tile: 1024 float4 chunks, 8 per thread
    const float*   gA   = feat + (size_t)tile * 16 * INFEAT;
    const unsigned ldsA = (unsigned)(uintptr_t)(void*)As;
    for (int i = tid; i < 16 * (INFEAT / 4); i += 128) {
        const int r = i >> 6, c4 = i & 63;                  // row, float4 chunk
        async_copy_b128(ldsA + (unsigned)(r * 260 + c4 * 4) * 4u,
                        gA, (unsigned)(r * INFEAT + c4 * 4) * 4u);
    }
    async_wait0();
    __syncthreads();

    const int cb = wave;                                    // output col tile
    const float* bRow = fcW + (size_t)(cb * 16 + l16) * INFEAT;
    const float* aRow = &As[l16 * 260];
    v8f acc = {0.f, 0.f, 0.f, 0.f, 0.f, 0.f, 0.f, 0.f};

    #pragma unroll 4
    for (int kk = 0; kk < INFEAT; kk += 4) {
        v2f a, b;
        a.x = aRow[kk + 2 * half];     a.y = aRow[kk + 2 * half + 1];
        b.x = bRow[kk + 2 * half];     b.y = bRow[kk + 2 * half + 1];
        acc = __builtin_amdgcn_wmma_f32_16x16x4_f32(false, a, false, b,
                                                    (short)0, acc, false, false);
    }

    float* out = feat_src + (size_t)tile * 16 * OUTF + cb * 16 + l16;
    #pragma unroll
    for (int j = 0; j < 8; ++j)
        out[(size_t)(j + 8 * half) * OUTF] = acc[j];
}

// ---------------------------------------------------------------------------
// K2: fused  ee = e_feat @ fce_W.T + b   and   e = sum(ee * feat_src[src], -1)
// Never materializes ee (saves ~800 MB of HBM traffic).
// block = 128 = 4 waves; 16-edge tile per block; wave h computes head h.
// e_feat tile async-staged with TH_LOAD_NT (read-once 410 MB stream; keep L2
// for the hot feat_src gather + softmax accumulators). After the 16 WMMAs,
// each half-wave row-reduces (acc+bias)*gather over 16 lanes via shfl_xor.
// ---------------------------------------------------------------------------
__global__ __launch_bounds__(128)
void k_edge_attn(const float* __restrict__ efeat,
                 const float* __restrict__ fceW,
                 const float* __restrict__ fceB,
                 const float* __restrict__ feat_src,
                 const int* __restrict__ src,
                 float* __restrict__ e_raw) {
    __shared__ float Es[16 * 68];            // padded pitch: 68 % 64 == 4
    __shared__ int   Ss[16];
    const int tile = blockIdx.x;
    const int tid  = threadIdx.x;
    const int wave = tid >> 5;               // == head h
    const int lane = tid & 31;
    const int half = lane >> 4;
    const int l16  = lane & 15;

    // async-stage the 16x64 e_feat tile: 256 float4 chunks, 2 per thread
    const float*   gE   = efeat + (size_t)tile * 16 * EFEAT;
    const unsigned ldsE = (unsigned)(uintptr_t)(void*)Es;
    for (int i = tid; i < 16 * (EFEAT / 4); i += 128) {
        const int r = i >> 4, c4 = i & 15;
        async_copy_b128_nt(ldsE + (unsigned)(r * 68 + c4 * 4) * 4u,
                           gE, (unsigned)(r * EFEAT + c4 * 4) * 4u);
    }
    if (tid < 16) Ss[tid] = src[tile * 16 + tid];
    async_wait0();
    __syncthreads();

    const int h = wave;
    const float* bRow = fceW + (size_t)(h * 16 + l16) * EFEAT;
    const float* aRow = &Es[l16 * 68];
    v8f acc = {0.f, 0.f, 0.f, 0.f, 0.f, 0.f, 0.f, 0.f};

    #pragma unroll
    for (int kk = 0; kk < EFEAT; kk += 4) {
        v2f a, b;
        a.x = aRow[kk + 2 * half];     a.y = aRow[kk + 2 * half + 1];
        b.x = bRow[kk + 2 * half];     b.y = bRow[kk + 2 * half + 1];
        acc = __builtin_amdgcn_wmma_f32_16x16x4_f32(false, a, false, b,
                                                    (short)0, acc, false, false);
    }

    const float bias = fceB[h * 16 + l16];
    #pragma unroll
    for (int j = 0; j < 8; ++j) {
        const int m    = j + 8 * half;                       // edge row in tile
        const int srow = Ss[m];
        const float el = feat_src[(size_t)srow * OUTF + h * 16 + l16];
        float p = (acc[j] + bias) * el;
        p += __shfl_xor(p, 1, 16);
        p += __shfl_xor(p, 2, 16);
        p += __shfl_xor(p, 4, 16);
        p += __shfl_xor(p, 8, 16);
        if (l16 == 0)
            e_raw[(size_t)(tile * 16 + m) * NH + h] = p;
    }
}

// ---------------------------------------------------------------------------
// K3: zero/neutral-init node accumulators (harness does not re-poison).
// ---------------------------------------------------------------------------
__global__ void k_init(unsigned* __restrict__ mkey, float* __restrict__ den,
                       float* __restrict__ attn, float* __restrict__ ft) {
    const int i = blockIdx.x * blockDim.x + threadIdx.x;
    if (i < NNODES * NH) { mkey[i] = 0u; den[i] = 0.f; attn[i] = 0.f; }
    if (i < NNODES * OUTF) ft[i] = 0.f;
}

// ---------------------------------------------------------------------------
// K4: segment max over dst via uint-keyed atomicMax (L2-resident, 0.8 MB).
// ---------------------------------------------------------------------------
__global__ void k_segmax(const float* __restrict__ e_raw,
                         const int* __restrict__ dst,
                         unsigned* __restrict__ mkey) {
    const int i = blockIdx.x * blockDim.x + threadIdx.x;
    if (i >= NEDGES * NH) return;
    const int e = i >> 2, h = i & 3;
    atomicMax(&mkey[dst[e] * NH + h], fkey(e_raw[i]));
}

// ---------------------------------------------------------------------------
// K5: num = exp(e - m[dst]); den[dst] += num; stash num in the `a` output.
// ---------------------------------------------------------------------------
__global__ void k_expden(const float* __restrict__ e_raw,
                         const int* __restrict__ dst,
                         const unsigned* __restrict__ mkey,
                         float* __restrict__ den,
                         float* __restrict__ a_out) {
    const int i = blockIdx.x * blockDim.x + threadIdx.x;
    if (i >= NEDGES * NH) return;
    const int e = i >> 2, h = i & 3;
    const int d = dst[e];
    const unsigned k = mkey[d * NH + h];
    float m = (k == 0u) ? 0.f : funkey(k);
    if (!isfinite(m)) m = 0.f;                  // mirror reference empty-seg rule
    const float num = expf(e_raw[i] - m);
    a_out[i] = num;
    atomicAdd(&den[d * NH + h], num);
}

// ---------------------------------------------------------------------------
// K6: a = num / den[dst]; attn_sum[dst] += a; ft[dst] += a * feat_src[src].
// One thread per (edge, head); 16 contiguous atomicAdds into ft (L2, 12.8 MB).
// ---------------------------------------------------------------------------
__global__ void k_norm_scatter(const int* __restrict__ src,
                               const int* __restrict__ dst,
                               const float* __restrict__ den,
                               const float* __restrict__ feat_src,
                               float* __restrict__ a_out,
                               float* __restrict__ attn,
                               float* __restrict__ ft) {
    const int i = blockIdx.x * blockDim.x + threadIdx.x;
    if (i >= NEDGES * NH) return;
    const int e = i >> 2, h = i & 3;
    const int d = dst[e];
    const float a = a_out[i] / den[d * NH + h];
    a_out[i] = a;
    atomicAdd(&attn[d * NH + h], a);
    const float* fs = feat_src + (size_t)src[e] * OUTF + h * HD;
    float* fo = ft + (size_t)d * OUTF + h * HD;
    #pragma unroll
    for (int dd = 0; dd < HD; ++dd)
        atomicAdd(&fo[dd], a * fs[dd]);
}

// ---------------------------------------------------------------------------
// K7: rst = ft / (attn_sum + 1e-12)
// ---------------------------------------------------------------------------
__global__ void k_final(const float* __restrict__ ft,
                        const float* __restrict__ attn,
                        float* __restrict__ rst) {
    const int i = blockIdx.x * blockDim.x + threadIdx.x;
    if (i >= NNODES * OUTF) return;
    const int n = i / OUTF;
    const int h = (i % OUTF) / HD;
    rst[i] = ft[i] / (attn[n * NH + h] + 1e-12f);
}

// ---------------------------------------------------------------------------
extern "C" void kernel_launch(void* const* d_in, const int* in_sizes, int n_in,
                              void* d_out, int out_size, void* d_ws, size_t ws_size,
                              hipStream_t stream) {
    const float* feat   = (const float*)d_in[0];
    const float* e_feat = (const float*)d_in[1];
    // d_in[2..5]: h0, attn_l, attn_r, attn_e — unused in the live forward path
    const int*   src    = (const int*)d_in[6];
    const int*   dst    = (const int*)d_in[7];
    const float* fcW    = (const float*)d_in[8];
    const float* fceW   = (const float*)d_in[9];
    const float* fceB   = (const float*)d_in[10];

    // workspace carve-up (floats): 13.4 M elems = 53.6 MB
    float*    feat_src = (float*)d_ws;                     // N*64
    float*    e_raw    = feat_src + (size_t)NNODES * OUTF; // E*4
    unsigned* mkey     = (unsigned*)(e_raw + (size_t)NEDGES * NH); // N*4
    float*    den      = (float*)(mkey + (size_t)NNODES * NH);     // N*4
    float*    attn     = den + (size_t)NNODES * NH;                // N*4
    float*    ft       = attn + (size_t)NNODES * NH;               // N*64

    float* rst   = (float*)d_out;                          // [N,H,D]
    float* a_out = rst + (size_t)NNODES * OUTF;            // [E,H,1]

    k_node_gemm<<<NNODES / 16, 128, 0, stream>>>(feat, fcW, feat_src);
    k_init<<<(NNODES * OUTF + 255) / 256, 256, 0, stream>>>(mkey, den, attn, ft);
    k_edge_attn<<<NEDGES / 16, 128, 0, stream>>>(e_feat, fceW, fceB, feat_src,
                                                 src, e_raw);
    k_segmax<<<(NEDGES * NH + 255) / 256, 256, 0, stream>>>(e_raw, dst, mkey);
    k_expden<<<(NEDGES * NH + 255) / 256, 256, 0, stream>>>(e_raw, dst, mkey,
                                                            den, a_out);
    k_norm_scatter<<<(NEDGES * NH + 255) / 256, 256, 0, stream>>>(src, dst, den,
                                                                  feat_src, a_out,
                                                                  attn, ft);
    k_final<<<(NNODES * OUTF + 255) / 256, 256, 0, stream>>>(ft, attn, rst);
}